// PositionalEncoder_41523743818357
// MI455X (gfx1250) — compile-verified
//
#include <hip/hip_runtime.h>
#include <math.h>

// out[b,s,d] = x[b,s,d] * sqrt(512) + pe(s,d)
//   pe(s,d) = sin(angle) if d even else cos(angle),
//   angle   = s / 10000^(2*d/512) = s * exp2(-d * 2*log2(1e4)/512)
//
// Shapes fixed by the reference: B=16, S=4096, D=512 (fp32 in/out).
// Pure streaming op: 268 MB of HBM traffic -> ~11.5 us floor at 23.3 TB/s.
// Strategy: B128 NT loads/stores, pe computed once per (s,d) column slot in
// registers and reused across the 16-deep batch loop (stride 8 MiB).

typedef float v4f __attribute__((ext_vector_type(4)));

#define D_MODEL   512
#define SEQ_LEN   4096
#define BATCH     16
#define ROWS_PER_BLOCK 2   // 128 threads per row (128 * float4 = 512 floats)

__global__ __launch_bounds__(256) void
PositionalEncoder_pe_stream_kernel(const float* __restrict__ x,
                                   float* __restrict__ out) {
    const int lane_in_row = threadIdx.x & 127;                 // 0..127
    const int row_in_blk  = threadIdx.x >> 7;                  // 0..1
    const int s = blockIdx.x * ROWS_PER_BLOCK + row_in_blk;    // 0..4095
    const int j = lane_in_row << 2;                            // d base, 0..508

    // inv_freq(d) = exp2(-d * 2*log2(10000)/512) ; lowers to v_exp_f32
    const float c   = 13.2877123795494f * (2.0f / (float)D_MODEL);
    const float pos = (float)s;

    const float a0 = pos * __builtin_amdgcn_exp2f(-c * (float)(j + 0));
    const float a1 = pos * __builtin_amdgcn_exp2f(-c * (float)(j + 1));
    const float a2 = pos * __builtin_amdgcn_exp2f(-c * (float)(j + 2));
    const float a3 = pos * __builtin_amdgcn_exp2f(-c * (float)(j + 3));

    // hardware v_sin/v_cos take input in revolutions: pre-scale by 1/(2*pi)
    const float inv2pi = 0.15915494309189535f;
    v4f pe;
    pe.x = __builtin_amdgcn_sinf(a0 * inv2pi);   // even d -> sin
    pe.y = __builtin_amdgcn_cosf(a1 * inv2pi);   // odd  d -> cos
    pe.z = __builtin_amdgcn_sinf(a2 * inv2pi);
    pe.w = __builtin_amdgcn_cosf(a3 * inv2pi);

    const float scale = 22.627416997969522f;  // sqrt(512)

    const size_t base    = (size_t)s * D_MODEL + (size_t)j;
    const size_t bstride = (size_t)SEQ_LEN * D_MODEL;   // 2M elems = 8 MiB

    const float* __restrict__ xp = x   + base;
    float*       __restrict__ op = out + base;

#pragma unroll
    for (int b = 0; b < BATCH; ++b) {
        const v4f* src = (const v4f*)(xp + (size_t)b * bstride);
        v4f*       dst = (v4f*)      (op + (size_t)b * bstride);

        v4f v = __builtin_nontemporal_load(src);   // global_load_b128 ... nt
        v4f r;
        r.x = fmaf(v.x, scale, pe.x);
        r.y = fmaf(v.y, scale, pe.y);
        r.z = fmaf(v.z, scale, pe.z);
        r.w = fmaf(v.w, scale, pe.w);
        __builtin_nontemporal_store(r, dst);       // global_store_b128 ... nt
    }
}

extern "C" void kernel_launch(void* const* d_in, const int* in_sizes, int n_in,
                              void* d_out, int out_size, void* d_ws, size_t ws_size,
                              hipStream_t stream) {
    (void)in_sizes; (void)n_in; (void)out_size; (void)d_ws; (void)ws_size;

    const float* x   = (const float*)d_in[0];
    float*       out = (float*)d_out;

    dim3 grid(SEQ_LEN / ROWS_PER_BLOCK);   // 2048 blocks
    dim3 block(256);                       // 8 wave32s; 2 rows per block
    PositionalEncoder_pe_stream_kernel<<<grid, block, 0, stream>>>(x, out);
}